// CNN_Flow_Layer_15032385536481
// MI455X (gfx1250) — compile-verified
//
#include <hip/hip_runtime.h>
#include <cstdint>

#define DIM 4096
#define BATCH 8192
#define THREADS 256
#define PER 16          // elements per thread: 256*16 = 4096
#define NEG_SLOPE 0.01f

typedef float v4f __attribute__((ext_vector_type(4)));

// ---------------------------------------------------------------------------
// Kernel 1: per-column precompute. scale[d], lpos[d], lneg[d] -> d_ws
//   scale = (w0==0) ? lmbd : (w0>0 ? -1/w0 + softplus(lmbd) : -1/w0 - softplus(lmbd))
//   lpos  = log|scale*w0 + 1|        (act_grad = 1 branch)
//   lneg  = log|0.01*scale*w0 + 1|   (act_grad = 0.01 branch)
// ---------------------------------------------------------------------------
__global__ __launch_bounds__(256) void flow_precompute(
    const float* __restrict__ conv_w, const float* __restrict__ lmbd,
    float* __restrict__ ws)
{
    int d = blockIdx.x * blockDim.x + threadIdx.x;
    if (d >= DIM) return;
    float w0 = conv_w[0];
    float l  = lmbd[d];
    // numerically stable softplus: max(l,0) + log1p(exp(-|l|))
    float sp = fmaxf(l, 0.0f) + log1pf(expf(-fabsf(l)));
    float scale = (w0 == 0.0f) ? l
                : ((w0 > 0.0f) ? (-1.0f / w0 + sp) : (-1.0f / w0 - sp));
    float lpos = logf(fabsf(fmaf(scale, w0, 1.0f)));
    float lneg = logf(fabsf(fmaf(NEG_SLOPE * scale, w0, 1.0f)));
    ws[d]           = scale;
    ws[DIM + d]     = lpos;
    ws[2 * DIM + d] = lneg;
}

// ---------------------------------------------------------------------------
// Kernel 2: one block per row. Async-stage the 16KB row into LDS (NT hint:
// x is read exactly once), apply the 3-tap stencil + leaky-ReLU + scale +
// skip (NT stores: out is never re-read), and reduce logdet (wave32).
// ---------------------------------------------------------------------------
__global__ __launch_bounds__(THREADS) void flow_main(
    const float* __restrict__ x,
    const float* __restrict__ conv_w, const float* __restrict__ conv_b,
    const float* __restrict__ ws,
    float* __restrict__ out, float* __restrict__ logdet)
{
    __shared__ float tile[DIM + 8];   // +8: zero pad for the right-padded stencil
    __shared__ float wred[THREADS / 32];

    const int tid = threadIdx.x;
    const int row = blockIdx.x;

    // zero the pad slots (only indices DIM, DIM+1 are read)
    if (tid < 8) tile[DIM + tid] = 0.0f;

    // ---- async global -> LDS staging (CDNA5 path, tracked by ASYNCcnt) ----
    // GVS mode: saddr = row base, vaddr = 32-bit byte offset, vdst = LDS byte addr
    // TH_LOAD_NT: x is streamed once; keep it out of L2 so the hot per-column
    // tables stay resident.
    const uint64_t rowbase = (uint64_t)(uintptr_t)(x + (size_t)row * DIM);
    const uint32_t ldsbase = (uint32_t)(uintptr_t)(&tile[0]);  // low 32 bits = LDS offset
#pragma unroll
    for (int q = 0; q < 4; ++q) {
        uint32_t off = (uint32_t)(q * THREADS + tid) * 16u;    // 16B chunks, coalesced
        uint32_t lds = ldsbase + off;
        asm volatile("global_load_async_to_lds_b128 %0, %1, %2 th:TH_LOAD_NT"
                     :: "v"(lds), "v"(off), "s"(rowbase) : "memory");
    }
    asm volatile("s_wait_asynccnt 0" ::: "memory");
    __syncthreads();

    // scalar params (uniform loads)
    const float w0 = conv_w[0], w1 = conv_w[1], w2 = conv_w[2];
    const float cb = conv_b[0];

    const int d0 = tid * PER;

    // x window: 16 outputs need xs[0..17]
    float xs[PER + 2];
    {
        const float4* t4 = (const float4*)(&tile[d0]);  // d0 is 16-float aligned
#pragma unroll
        for (int q = 0; q < 4; ++q) {
            float4 v = t4[q];
            xs[4 * q + 0] = v.x; xs[4 * q + 1] = v.y;
            xs[4 * q + 2] = v.z; xs[4 * q + 3] = v.w;
        }
        xs[PER]     = tile[d0 + PER];
        xs[PER + 1] = tile[d0 + PER + 1];
    }

    // column tables (48 KB total, L2-resident, reused by all 8192 blocks)
    float scv[PER], lpv[PER], lnv[PER];
    {
        const float4* s4 = (const float4*)(ws + d0);
        const float4* p4 = (const float4*)(ws + DIM + d0);
        const float4* n4 = (const float4*)(ws + 2 * DIM + d0);
#pragma unroll
        for (int q = 0; q < 4; ++q) {
            float4 s = s4[q], p = p4[q], n = n4[q];
            scv[4 * q + 0] = s.x; scv[4 * q + 1] = s.y; scv[4 * q + 2] = s.z; scv[4 * q + 3] = s.w;
            lpv[4 * q + 0] = p.x; lpv[4 * q + 1] = p.y; lpv[4 * q + 2] = p.z; lpv[4 * q + 3] = p.w;
            lnv[4 * q + 0] = n.x; lnv[4 * q + 1] = n.y; lnv[4 * q + 2] = n.z; lnv[4 * q + 3] = n.w;
        }
    }

    // stencil + activation + scale + skip; logdet via table select
    float acc = 0.0f;
    float ov[PER];
#pragma unroll
    for (int j = 0; j < PER; ++j) {
        float c = fmaf(xs[j], w0, fmaf(xs[j + 1], w1, fmaf(xs[j + 2], w2, cb)));
        bool pos = (c >= 0.0f);
        float act = pos ? c : NEG_SLOPE * c;
        ov[j] = fmaf(act, scv[j], xs[j]);
        acc += pos ? lpv[j] : lnv[j];
    }

    // store 16 contiguous outputs as 4x b128, non-temporal (never re-read)
    {
        v4f* o4 = (v4f*)(out + (size_t)row * DIM + d0);
#pragma unroll
        for (int q = 0; q < 4; ++q) {
            v4f v = { ov[4 * q + 0], ov[4 * q + 1], ov[4 * q + 2], ov[4 * q + 3] };
            __builtin_nontemporal_store(v, &o4[q]);
        }
    }

    // ---- logdet reduction: wave32 shuffle tree, then cross-wave via LDS ----
#pragma unroll
    for (int o = 16; o > 0; o >>= 1)
        acc += __shfl_xor(acc, o, 32);
    const int wave = tid >> 5;
    const int lane = tid & 31;
    if (lane == 0) wred[wave] = acc;
    __syncthreads();
    if (tid == 0) {
        float s = 0.0f;
#pragma unroll
        for (int i = 0; i < THREADS / 32; ++i) s += wred[i];
        logdet[row] = s;
    }
}

// ---------------------------------------------------------------------------
extern "C" void kernel_launch(void* const* d_in, const int* in_sizes, int n_in,
                              void* d_out, int out_size, void* d_ws, size_t ws_size,
                              hipStream_t stream) {
    const float* x      = (const float*)d_in[0];  // [BATCH, DIM]
    const float* conv_w = (const float*)d_in[1];  // [1,1,3]
    const float* conv_b = (const float*)d_in[2];  // [1]
    const float* lmbd   = (const float*)d_in[3];  // [DIM]

    float* ws     = (float*)d_ws;                 // 3*DIM floats (48 KB)
    float* out    = (float*)d_out;                // [BATCH, DIM]
    float* logdet = out + (size_t)BATCH * DIM;    // [BATCH]

    flow_precompute<<<DIM / 256, 256, 0, stream>>>(conv_w, lmbd, ws);
    flow_main<<<BATCH, THREADS, 0, stream>>>(x, conv_w, conv_b, ws, out, logdet);
}